// GAT_76922864272073
// MI455X (gfx1250) — compile-verified
//
#include <hip/hip_runtime.h>
#include <math.h>
#include <stdint.h>

// ---------------------------------------------------------------------------
// GAT (3-layer) forward for MI455X / gfx1250, wave32.
//   N=100000 nodes, E=1600000 edges (+N self loops), HEADS=2.
//   Layer0: 128 -> 2x32 (concat->64), ELU
//   Layer1:  64 -> 2x32 (concat->64), ELU
//   Layer2:  64 -> 2x64 (mean ->64), log_softmax
// Dense projections: V_WMMA_F32_16X16X4_F32 (exact fp32 matrix pipe) with the
// W column-slice staged into LDS by the Tensor Data Mover (tensor_load_to_lds,
// TENSORcnt). Edge softmax/aggregation: scatter-atomic passes (L2-resident).
// ---------------------------------------------------------------------------

#define GAT_N   100000
#define GAT_E   1600000
#define GAT_H   2
#define NEG_SLOPE 0.2f

typedef float v2f  __attribute__((ext_vector_type(2)));
typedef float v8f  __attribute__((ext_vector_type(8)));
typedef unsigned int u32x4 __attribute__((ext_vector_type(4)));
typedef int   i32x4 __attribute__((ext_vector_type(4)));
typedef int   i32x8 __attribute__((ext_vector_type(8)));

// float atomic max via signed-max / unsigned-min trick (monotone under IEEE bits)
__device__ __forceinline__ void atomicMaxF32(float* addr, float v) {
  if (v >= 0.0f) atomicMax((int*)addr, __float_as_int(v));
  else           atomicMin((unsigned int*)addr, __float_as_uint(v));
}

// ---------------------------------------------------------------------------
// GEMM: Hout[nRows x nCols] = X[nRows x K] @ W[K x nCols], row-major.
// Block = 256 threads (8 waves): 8 consecutive 16-row tiles x one 16-col tile.
// The K x 16 slice of W for this column tile is DMA'd to LDS once by the TDM
// (wave 0 issues tensor_load_to_lds, waits s_wait_tensorcnt 0, block barrier),
// then all 8 waves feed WMMA's B operand from LDS.
//
// WMMA fp32 lane layouts (ISA 7.12.2):
//   A (16x4): lane<16 -> row=lane, K+0/K+1 ; lane>=16 -> row=lane-16, K+2/K+3.
//   B (4x16): vgpr0 = K+0 (lanes 0-15) / K+2 (lanes 16-31); vgpr1 = K+1 / K+3.
//   C/D(16x16): vgpr r -> M = r + (lane<16 ? 0 : 8), N = lane&15.
// ---------------------------------------------------------------------------
template <int K>
__global__ void gat_gemm_wmma(const float* __restrict__ X, const float* __restrict__ W,
                              float* __restrict__ Hout, int nRows, int nCols) {
  __shared__ float shW[K * 16];                  // K rows x 16 cols, <= 8 KB

  const int lane    = threadIdx.x & 31;
  const int waveIdx = threadIdx.x >> 5;
  const int colBase = blockIdx.y << 4;

  // ---- TDM: stage W[:, colBase:colBase+16] into LDS (wave 0 only; EXEC
  // ignored by tensor ops, branch keeps other waves from re-issuing the DMA).
  if (waveIdx == 0) {
    const uint64_t gaddr = (uint64_t)(uintptr_t)(W + colBase);   // tile start
    const uint32_t ldsaddr = (uint32_t)(uintptr_t)(void*)&shW[0]; // LDS addr = low 32 bits

    u32x4 g0;
    g0[0] = 1u;                                        // count=1 (valid user D#)
    g0[1] = ldsaddr;                                   // lds_addr [63:32]
    g0[2] = (uint32_t)gaddr;                           // global_addr [95:64]
    g0[3] = (uint32_t)((gaddr >> 32) & 0x01FFFFFFu)    // global_addr [120:96]
          | (2u << 30);                                // type=2 ("image")

    i32x8 g1;
    g1[0] = 0x00020000;                                // data_size=2 -> 4 bytes
    g1[1] = (int)((nCols & 0xFFFF) << 16);             // tensor_dim0[15:0] @ [63:48]
    g1[2] = (int)(((unsigned)nCols >> 16) |            // tensor_dim0[31:16]
                  ((unsigned)K << 16));                // tensor_dim1[15:0]
    g1[3] = (int)(16u << 16);                          // tensor_dim1[31:16]=0, tile_dim0=16
    g1[4] = (int)(unsigned)K;                          // tile_dim1=K, tile_dim2=0
    g1[5] = nCols;                                     // tensor_dim0_stride[31:0]
    g1[6] = 0;                                         // stride hi / dim1_stride lo
    g1[7] = 0;

    i32x4 zero4 = {0, 0, 0, 0};                        // groups 2/3 unused (2D tile)
    i32x8 zero8 = {0, 0, 0, 0, 0, 0, 0, 0};
    __builtin_amdgcn_tensor_load_to_lds(g0, g1, zero4, zero4, zero8, 0);
    __builtin_amdgcn_s_wait_tensorcnt(0);
  }
  __syncthreads();

  const int rowTile = blockIdx.x * 8 + waveIdx;
  if (rowTile < (nRows >> 4)) {                  // wave-uniform: EXEC stays all-ones
    const int rowBase = rowTile << 4;
    const int half = lane >> 4;                  // 0 for lanes 0-15, 1 for 16-31
    const int l16  = lane & 15;

    const float* xrow = X + (size_t)(rowBase + l16) * K;
    v8f acc = {};
#pragma unroll
    for (int k0 = 0; k0 < K; k0 += 4) {
      const int ka = k0 + 2 * half;
      v2f a, b;
      a.x = xrow[ka + 0];
      a.y = xrow[ka + 1];
      b.x = shW[(ka + 0) * 16 + l16];            // ds_load from TDM-staged slice
      b.y = shW[(ka + 1) * 16 + l16];
      acc = __builtin_amdgcn_wmma_f32_16x16x4_f32(false, a, false, b,
                                                  (short)0, acc, false, false);
    }
    const int outCol = colBase + l16;
    const int rowOff = rowBase + (half ? 8 : 0);
#pragma unroll
    for (int r = 0; r < 8; ++r)
      Hout[(size_t)(rowOff + r) * nCols + outCol] = acc[r];
  }
}

// alpha_s[n,h] = sum_c h[n,h,c]*a_src[h,c]  (and same for a_dst)
__global__ void gat_alpha(const float* __restrict__ Hb, const float* __restrict__ avs,
                          const float* __restrict__ avd, float* __restrict__ asb,
                          float* __restrict__ adb, int n, int C) {
  const int t = blockIdx.x * blockDim.x + threadIdx.x;
  if (t >= n * GAT_H) return;
  const int node = t / GAT_H, h = t % GAT_H;
  const float* hp = Hb + (size_t)node * GAT_H * C + (size_t)h * C;
  float s = 0.f, d = 0.f;
  for (int c = 0; c < C; ++c) {
    const float v = hp[c];
    s += v * avs[h * C + c];
    d += v * avd[h * C + c];
  }
  asb[t] = s;
  adb[t] = d;
}

__global__ void fill_f32(float* __restrict__ p, float v, int n) {
  const int t = blockIdx.x * blockDim.x + threadIdx.x;
  if (t < n) p[t] = v;
}

// pass 1: logits + segment-max(dst)
__global__ void edge_logit_max(const int* __restrict__ ei, const float* __restrict__ asb,
                               const float* __restrict__ adb, float* __restrict__ logit,
                               float* __restrict__ mb) {
  const int e = blockIdx.x * blockDim.x + threadIdx.x;
  const int Etot = GAT_E + GAT_N;
  if (e >= Etot) return;
  int s, d;
  if (e < GAT_E) { s = ei[e]; d = ei[GAT_E + e]; }
  else           { s = d = e - GAT_E; }
#pragma unroll
  for (int h = 0; h < GAT_H; ++h) {
    float x = asb[s * GAT_H + h] + adb[d * GAT_H + h];
    x = (x > 0.f) ? x : NEG_SLOPE * x;
    logit[(size_t)e * GAT_H + h] = x;
    atomicMaxF32(&mb[d * GAT_H + h], x);
  }
}

// pass 2: e = exp(logit - m[dst]) (stored in place) + segment-sum(dst)
__global__ void edge_exp_sum(const int* __restrict__ ei, const float* __restrict__ mb,
                             float* __restrict__ ebuf, float* __restrict__ denom) {
  const int e = blockIdx.x * blockDim.x + threadIdx.x;
  const int Etot = GAT_E + GAT_N;
  if (e >= Etot) return;
  const int d = (e < GAT_E) ? ei[GAT_E + e] : (e - GAT_E);
#pragma unroll
  for (int h = 0; h < GAT_H; ++h) {
    const float ex = expf(ebuf[(size_t)e * GAT_H + h] - mb[d * GAT_H + h]);
    ebuf[(size_t)e * GAT_H + h] = ex;
    atomicAdd(&denom[d * GAT_H + h], ex);
  }
}

// pass 3: agg[dst] += h[src] * alpha  — one wave per edge, lanes stride channels
__global__ void edge_aggregate(const int* __restrict__ ei, const float* __restrict__ Hb,
                               const float* __restrict__ ebuf, const float* __restrict__ denom,
                               float* __restrict__ agg, int C) {
  const int t = blockIdx.x * blockDim.x + threadIdx.x;
  const int wave = t >> 5, lane = t & 31;
  const int Etot = GAT_E + GAT_N;
  if (wave >= Etot) return;
  int s, d;
  if (wave < GAT_E) { s = ei[wave]; d = ei[GAT_E + wave]; }
  else              { s = d = wave - GAT_E; }
  const int HC = GAT_H * C;
  const float* hsrc = Hb + (size_t)s * HC;
  float* aggd = agg + (size_t)d * HC;
  __builtin_prefetch(hsrc, 0, 3);  // -> global_prefetch_b8
  for (int idx = lane; idx < HC; idx += 32) {
    const int h = idx / C;
    const float alpha = ebuf[(size_t)wave * GAT_H + h] / (denom[d * GAT_H + h] + 1e-16f);
    atomicAdd(&aggd[idx], hsrc[idx] * alpha);
  }
}

// concat layers: y = elu(agg + b)
__global__ void epilogue_elu(const float* __restrict__ agg, const float* __restrict__ bv,
                             float* __restrict__ xout, int total, int HC) {
  const int t = blockIdx.x * blockDim.x + threadIdx.x;
  if (t >= total) return;
  const float v = agg[t] + bv[t % HC];
  xout[t] = (v > 0.f) ? v : (expf(v) - 1.f);
}

// final layer: mean over 2 heads (C=64) + bias, then log_softmax; one wave per node
__global__ void epilogue_final(const float* __restrict__ agg, const float* __restrict__ bv,
                               float* __restrict__ out, int Nn) {
  const int t = blockIdx.x * blockDim.x + threadIdx.x;
  const int node = t >> 5, lane = t & 31;
  if (node >= Nn) return;
  const float* a0 = agg + (size_t)node * 128;            // [head0(64) | head1(64)]
  float v0 = 0.5f * (a0[lane]      + a0[64 + lane])      + bv[lane];
  float v1 = 0.5f * (a0[lane + 32] + a0[64 + lane + 32]) + bv[lane + 32];
  float mx = fmaxf(v0, v1);
  for (int off = 16; off > 0; off >>= 1) mx = fmaxf(mx, __shfl_xor(mx, off));
  float se = expf(v0 - mx) + expf(v1 - mx);
  for (int off = 16; off > 0; off >>= 1) se += __shfl_xor(se, off);
  const float lse = mx + logf(se);
  out[(size_t)node * 64 + lane]      = v0 - lse;
  out[(size_t)node * 64 + lane + 32] = v1 - lse;
}

// ---------------------------------------------------------------------------
template <int K>
static void run_layer(const float* xin, int C, bool concat,
                      const float* W, const float* avs, const float* avd, const float* bv,
                      const int* ei, float* hbuf, float* agg, float* asb, float* adb,
                      float* mb, float* db, float* eb, float* xout, hipStream_t stream) {
  const int N = GAT_N, Etot = GAT_E + GAT_N;
  const int HC = GAT_H * C;

  const int rowTiles = N / 16;                         // 6250
  dim3 gemmGrid((rowTiles + 7) / 8, HC / 16);
  gat_gemm_wmma<K><<<gemmGrid, 256, 0, stream>>>(xin, W, hbuf, N, HC);

  gat_alpha<<<(N * GAT_H + 255) / 256, 256, 0, stream>>>(hbuf, avs, avd, asb, adb, N, C);

  fill_f32<<<(N * GAT_H + 255) / 256, 256, 0, stream>>>(mb, -INFINITY, N * GAT_H);
  fill_f32<<<(N * GAT_H + 255) / 256, 256, 0, stream>>>(db, 0.f, N * GAT_H);
  fill_f32<<<(N * HC + 255) / 256, 256, 0, stream>>>(agg, 0.f, N * HC);

  edge_logit_max<<<(Etot + 255) / 256, 256, 0, stream>>>(ei, asb, adb, eb, mb);
  edge_exp_sum<<<(Etot + 255) / 256, 256, 0, stream>>>(ei, mb, eb, db);
  edge_aggregate<<<(unsigned)(((size_t)Etot * 32 + 255) / 256), 256, 0, stream>>>(
      ei, hbuf, eb, db, agg, C);

  if (concat)
    epilogue_elu<<<(N * HC + 255) / 256, 256, 0, stream>>>(agg, bv, xout, N * HC, HC);
  else
    epilogue_final<<<(N * 32 + 255) / 256, 256, 0, stream>>>(agg, bv, xout, N);
}

extern "C" void kernel_launch(void* const* d_in, const int* in_sizes, int n_in,
                              void* d_out, int out_size, void* d_ws, size_t ws_size,
                              hipStream_t stream) {
  const int N = GAT_N;
  const float* x   = (const float*)d_in[0];
  const int*   ei  = (const int*)d_in[1];     // edge_index [2,E] (per harness: int)
  const float* W0  = (const float*)d_in[2];
  const float* as0 = (const float*)d_in[3];
  const float* ad0 = (const float*)d_in[4];
  const float* b0  = (const float*)d_in[5];
  const float* W1  = (const float*)d_in[6];
  const float* as1 = (const float*)d_in[7];
  const float* ad1 = (const float*)d_in[8];
  const float* b1  = (const float*)d_in[9];
  const float* W2  = (const float*)d_in[10];
  const float* as2 = (const float*)d_in[11];
  const float* ad2 = (const float*)d_in[12];
  const float* b2  = (const float*)d_in[13];

  // workspace layout (floats); total ~36.2M floats (~145 MB)
  float* ws   = (float*)d_ws;
  float* hbuf = ws;                               // N*128
  float* agg  = hbuf + (size_t)N * 128;           // N*128
  float* xbuf = agg  + (size_t)N * 128;           // N*64
  float* asb  = xbuf + (size_t)N * 64;            // N*2
  float* adb  = asb  + (size_t)N * GAT_H;         // N*2
  float* mb   = adb  + (size_t)N * GAT_H;         // N*2
  float* db   = mb   + (size_t)N * GAT_H;         // N*2
  float* eb   = db   + (size_t)N * GAT_H;         // (E+N)*2

  // Layer 0: 128 -> 2x32 concat, ELU
  run_layer<128>(x,    32, true,  W0, as0, ad0, b0, ei, hbuf, agg, asb, adb, mb, db, eb, xbuf, stream);
  // Layer 1: 64 -> 2x32 concat, ELU (in-place xbuf is safe: stream-ordered)
  run_layer<64>(xbuf,  32, true,  W1, as1, ad1, b1, ei, hbuf, agg, asb, adb, mb, db, eb, xbuf, stream);
  // Layer 2: 64 -> 2x64 mean, log_softmax -> d_out
  run_layer<64>(xbuf,  64, false, W2, as2, ad2, b2, ei, hbuf, agg, asb, adb, mb, db, eb,
                (float*)d_out, stream);
}